// MatchLSTM_33492154974231
// MI455X (gfx1250) — compile-verified
//
#include <hip/hip_runtime.h>
#include <math.h>

// ---------------------------------------------------------------------------
// MatchLSTM for MI455X (gfx1250, wave32).
// GEMMs: V_WMMA_F32_16X16X4_F32 with 2x2 register blocking per wave,
// A tiles staged via GLOBAL_LOAD_ASYNC_TO_LDS_B128 when available.
// ---------------------------------------------------------------------------

typedef __attribute__((ext_vector_type(2))) float v2f;
typedef __attribute__((ext_vector_type(8))) float v8f;
typedef __attribute__((ext_vector_type(4))) int   v4i;

static constexpr int Bq  = 128;   // batch
static constexpr int Pq  = 64;    // premise length
static constexpr int Kq  = 64;    // hypothesis length
static constexpr int Eq  = 300;   // embed dim
static constexpr int Hq  = 300;   // hidden dim
static constexpr int G4H = 1200;  // 4*H gate width
static constexpr int Cq  = 3;     // classes

#if defined(__HIP_DEVICE_COMPILE__) && __has_builtin(__builtin_amdgcn_global_load_async_to_lds_b128)
#define HAVE_ASYNC_LDS 1
typedef __attribute__((address_space(1))) v4i* gv4i_ptr;   // global int4*
typedef __attribute__((address_space(3))) v4i* lv4i_ptr;   // LDS int4*
#else
#define HAVE_ASYNC_LDS 0
#endif

// ------------------------- WMMA GEMM: D = X @ W^T (+bias) ------------------
// X: [M,K] row-major (ldx), W: [N,K] row-major (ldw)  => B[k][n] = W[n][k]
// Block: 256 threads = 8 waves (4 x 2). Wave tile 32x32 (2x2 of 16x16 WMMA).
// Block tile 128x64, K staged in 32-wide chunks.
#define BMT 128
#define BNT 64
#define KC  32
#define ASTR 36   // A LDS row stride: 144B (16B aligned for B128 async,
                  // r*36 mod 64 distinct for r=0..15 -> conflict-free)

__global__ __launch_bounds__(256) void wmma_gemm_xwt(
    const float* __restrict__ X, const float* __restrict__ W,
    const float* __restrict__ bias, float* __restrict__ D,
    int M, int N, int K, int ldx, int ldw, int ldd, int add_bias)
{
    __shared__ float As[BMT * ASTR];   // 18.0 KB
    __shared__ float Bs[KC * BNT];     //  8.0 KB

    const int tid  = threadIdx.x;
    const int wave = tid >> 5;         // 0..7
    const int lane = tid & 31;
    const int wm   = wave >> 1;        // 0..3 : 32-row strip in block tile
    const int wn   = wave & 1;         // 0..1 : 32-col strip in block tile
    const int l15  = lane & 15;
    const int half = lane >> 4;        // 0: K=kk,kk+1 ; 1: K=kk+2,kk+3
    const int blockM = blockIdx.x * BMT;
    const int blockN = blockIdx.y * BNT;

    v8f acc00 = {0.f,0.f,0.f,0.f,0.f,0.f,0.f,0.f};
    v8f acc01 = acc00, acc10 = acc00, acc11 = acc00;

    for (int k0 = 0; k0 < K; k0 += KC) {
        const bool fullChunk = (k0 + KC <= K) && (blockM + BMT <= M);
#if HAVE_ASYNC_LDS
        if (fullChunk) {
            // 128 rows x 32 floats = 1024 x 16B segments; 4 async B128 per thread
            #pragma unroll
            for (int t = 0; t < 4; ++t) {
                int i   = tid + t * 256;
                int r   = i >> 3;        // row 0..127
                int seg = i & 7;         // 16B segment within the 128B row
                __builtin_amdgcn_global_load_async_to_lds_b128(
                    (gv4i_ptr)(X + (size_t)(blockM + r) * ldx + k0 + seg * 4),
                    (lv4i_ptr)(As + r * ASTR + seg * 4),
                    0, 0);
            }
        } else
#endif
        {
            for (int i = tid; i < BMT * KC; i += 256) {
                int r = i >> 5, c = i & 31;                  // KC == 32
                int gm = blockM + r, gk = k0 + c;
                As[r * ASTR + c] = (gm < M && gk < K) ? X[(size_t)gm * ldx + gk] : 0.f;
            }
        }
        // B tile from W[N,K] (transposed gather)
        for (int i = tid; i < KC * BNT; i += 256) {
            int c = i >> 6, n = i & 63;                      // BNT == 64
            int gk = k0 + c, gn = blockN + n;
            Bs[c * BNT + n] = (gn < N && gk < K) ? W[(size_t)gn * ldw + gk] : 0.f;
        }
        // prefetch next A chunk (global_prefetch_b8)
        if (k0 + KC < K) {
            int gm = blockM + (tid >> 1);
            if (gm < M) __builtin_prefetch(&X[(size_t)gm * ldx + k0 + KC], 0, 1);
        }
#if HAVE_ASYNC_LDS
        asm volatile("s_wait_asynccnt 0" ::: "memory");
#endif
        __syncthreads();

        const int ra0 = (wm * 32 + l15) * ASTR;   // A rows for m-tile 0
        const int ra1 = ra0 + 16 * ASTR;          // A rows for m-tile 1
        const int cb  = wn * 32 + l15;            // B cols for n-tile 0
        #pragma unroll
        for (int kk = 0; kk < KC; kk += 4) {
            const int ka = kk + 2 * half;
            v2f a0, a1, b0, b1;
            a0.x = As[ra0 + ka];           a0.y = As[ra0 + ka + 1];
            a1.x = As[ra1 + ka];           a1.y = As[ra1 + ka + 1];
            b0.x = Bs[ka * BNT + cb];      b0.y = Bs[(ka + 1) * BNT + cb];
            b1.x = Bs[ka * BNT + cb + 16]; b1.y = Bs[(ka + 1) * BNT + cb + 16];
            acc00 = __builtin_amdgcn_wmma_f32_16x16x4_f32(false, a0, false, b0, (short)0, acc00, false, false);
            acc01 = __builtin_amdgcn_wmma_f32_16x16x4_f32(false, a0, false, b1, (short)0, acc01, false, false);
            acc10 = __builtin_amdgcn_wmma_f32_16x16x4_f32(false, a1, false, b0, (short)0, acc10, false, false);
            acc11 = __builtin_amdgcn_wmma_f32_16x16x4_f32(false, a1, false, b1, (short)0, acc11, false, false);
        }
        __syncthreads();
    }

    // store 2x2 tiles; D 16x16 layout: VGPR r -> M=r (lanes 0-15), M=r+8 (16-31)
    const int mBase = blockM + wm * 32;
    const int nBase = blockN + wn * 32;
    #pragma unroll
    for (int tn = 0; tn < 2; ++tn) {
        int n = nBase + tn * 16 + l15;
        if (n >= N) continue;
        float bv = add_bias ? bias[n] : 0.f;
        #pragma unroll
        for (int tm = 0; tm < 2; ++tm) {
            const v8f& acc = (tm == 0) ? (tn == 0 ? acc00 : acc01)
                                       : (tn == 0 ? acc10 : acc11);
            #pragma unroll
            for (int r = 0; r < 8; ++r) {
                int m = mBase + tm * 16 + r + 8 * half;
                if (m < M) D[(size_t)m * ldd + n] = acc[r] + bv;
            }
        }
    }
}

// ------------------------------ small kernels ------------------------------

__global__ void gather_embed(const int* __restrict__ tok,
                             const float* __restrict__ embed,
                             float* __restrict__ out, int rows)
{
    int idx = blockIdx.x * blockDim.x + threadIdx.x;
    int total = rows * Eq;
    if (idx >= total) return;
    int r = idx / Eq, e = idx - r * Eq;
    out[idx] = embed[(size_t)tok[r] * Eq + e];
}

__global__ void vec_add2(const float* __restrict__ a, const float* __restrict__ b,
                         float* __restrict__ o, int n)
{
    int i = blockIdx.x * blockDim.x + threadIdx.x;
    if (i < n) o[i] = a[i] + b[i];
}

__global__ void zero_f32(float* __restrict__ p, int n)
{
    int i = blockIdx.x * blockDim.x + threadIdx.x;
    if (i < n) p[i] = 0.f;
}

__device__ __forceinline__ float sigmf(float x) { return 1.f / (1.f + expf(-x)); }

// gates [rows,4H] (i,f,g,o) with zero prev state: h = sig(o)*tanh(sig(i)*tanh(g))
__global__ void lstm_act(const float* __restrict__ Gt, float* __restrict__ Hh, int rows)
{
    int idx = blockIdx.x * blockDim.x + threadIdx.x;
    int total = rows * Hq;
    if (idx >= total) return;
    int m = idx / Hq, j = idx - m * Hq;
    const float* g = Gt + (size_t)m * G4H;
    float gi = g[j];
    float gg = g[2 * Hq + j];
    float go = g[3 * Hq + j];
    float c  = sigmf(gi) * tanhf(gg);
    Hh[idx]  = sigmf(go) * tanhf(c);
}

// One block per batch row: masked attention over premise + build m = [a ; h_t_k]
__global__ __launch_bounds__(256) void attn_step(
    const float* __restrict__ s_proj,   // [B*P, H]
    const float* __restrict__ t_proj,   // [B*K, H]
    const float* __restrict__ m_proj,   // [B, H]
    const float* __restrict__ h_s,      // [B*P, H]
    const float* __restrict__ h_t,      // [B*K, H]
    const float* __restrict__ w_e,      // [H]
    const int*   __restrict__ plen,     // [B]
    float* __restrict__ m_cat,          // [B, 2H]
    int k)
{
    __shared__ float tm[Hq];
    __shared__ float es[Pq];
    const int b = blockIdx.x;
    const int tid = threadIdx.x;
    const int wave = tid >> 5, lane = tid & 31;

    for (int j = tid; j < Hq; j += 256)
        tm[j] = t_proj[((size_t)b * Kq + k) * Hq + j] + m_proj[(size_t)b * Hq + j];
    __syncthreads();

    const int pl = plen[b];
    for (int p = wave; p < Pq; p += 8) {
        const float* sp = s_proj + ((size_t)b * Pq + p) * Hq;
        float s = 0.f;
        for (int j = lane; j < Hq; j += 32)
            s += w_e[j] * tanhf(sp[j] + tm[j]);
        for (int off = 16; off; off >>= 1) s += __shfl_xor(s, off, 32);
        if (lane == 0) es[p] = (p < pl) ? s : -3.402823466e38f;
    }
    __syncthreads();

    if (tid == 0) {   // 64-entry softmax: negligible vs GEMMs
        float mx = -3.402823466e38f;
        for (int p = 0; p < Pq; ++p) mx = fmaxf(mx, es[p]);
        float sum = 0.f;
        for (int p = 0; p < Pq; ++p) { float e = expf(es[p] - mx); es[p] = e; sum += e; }
        float inv = 1.f / sum;
        for (int p = 0; p < Pq; ++p) es[p] *= inv;
    }
    __syncthreads();

    const float* hsb = h_s + (size_t)b * Pq * Hq;
    const float* htk = h_t + ((size_t)b * Kq + k) * Hq;
    for (int j = tid; j < Hq; j += 256) {
        float s = 0.f;
        for (int p = 0; p < Pq; ++p) s += es[p] * hsb[(size_t)p * Hq + j];
        m_cat[(size_t)b * (2 * Hq) + j]      = s;
        m_cat[(size_t)b * (2 * Hq) + Hq + j] = htk[j];
    }
}

// Gather h at last valid hypothesis step, FC [3,H], softmax over 3 classes.
__global__ void final_softmax(const float* __restrict__ hs_all,  // [K,B,H]
                              const int*   __restrict__ hlen,
                              const float* __restrict__ fc_w,    // [3,H]
                              const float* __restrict__ fc_b,    // [3]
                              float* __restrict__ out)           // [B,3]
{
    const int b = blockIdx.x;
    const int lane = threadIdx.x;
    int idx = hlen[b] - 1;
    if (idx < 0) idx = 0;
    if (idx >= Kq) idx = Kq - 1;
    const float* hf = hs_all + ((size_t)idx * Bq + b) * Hq;
    float lg[Cq];
    #pragma unroll
    for (int c = 0; c < Cq; ++c) {
        float s = 0.f;
        for (int j = lane; j < Hq; j += 32) s += hf[j] * fc_w[c * Hq + j];
        for (int off = 16; off; off >>= 1) s += __shfl_xor(s, off, 32);
        lg[c] = s + fc_b[c];
    }
    if (lane == 0) {
        float mx = fmaxf(lg[0], fmaxf(lg[1], lg[2]));
        float e0 = expf(lg[0] - mx), e1 = expf(lg[1] - mx), e2 = expf(lg[2] - mx);
        float inv = 1.f / (e0 + e1 + e2);
        out[b * Cq + 0] = e0 * inv;
        out[b * Cq + 1] = e1 * inv;
        out[b * Cq + 2] = e2 * inv;
    }
}

// ------------------------------- host driver -------------------------------

extern "C" void kernel_launch(void* const* d_in, const int* in_sizes, int n_in,
                              void* d_out, int out_size, void* d_ws, size_t ws_size,
                              hipStream_t stream)
{
    (void)in_sizes; (void)n_in; (void)out_size; (void)ws_size;

    const int*   premise        = (const int*)  d_in[0];
    const int*   premise_len    = (const int*)  d_in[1];
    const int*   hypothesis     = (const int*)  d_in[2];
    const int*   hypothesis_len = (const int*)  d_in[3];
    const float* embed          = (const float*)d_in[4];
    const float* w_e            = (const float*)d_in[5];
    const float* W_s            = (const float*)d_in[6];
    const float* W_t            = (const float*)d_in[7];
    const float* W_m            = (const float*)d_in[8];
    const float* fc_w           = (const float*)d_in[9];
    const float* fc_b           = (const float*)d_in[10];
    const float* Wih_p          = (const float*)d_in[11];
    const float* bih_p          = (const float*)d_in[13];
    const float* bhh_p          = (const float*)d_in[14];
    const float* Wih_h          = (const float*)d_in[15];
    const float* bih_h          = (const float*)d_in[17];
    const float* bhh_h          = (const float*)d_in[18];
    const float* Wih_m          = (const float*)d_in[19];
    const float* bih_m          = (const float*)d_in[21];
    const float* bhh_m          = (const float*)d_in[22];

    // carve workspace (256B aligned so B128 async global addresses stay 16B aligned)
    char* ws = (char*)d_ws;
    auto carve = [&](size_t bytes) -> float* {
        char* p = ws;
        ws += (bytes + 255) & ~(size_t)255;
        return (float*)p;
    };
    const size_t tokrows = (size_t)Bq * Pq;             // == Bq*Kq
    float* xbuf   = carve(sizeof(float) * tokrows * Eq);        // gathered embeds
    float* gates  = carve(sizeof(float) * tokrows * G4H);       // gate buffer (reused)
    float* h_s    = carve(sizeof(float) * tokrows * Hq);
    float* h_t    = carve(sizeof(float) * tokrows * Hq);
    float* s_proj = carve(sizeof(float) * tokrows * Hq);
    float* t_proj = carve(sizeof(float) * tokrows * Hq);
    float* hs_all = carve(sizeof(float) * (size_t)Kq * Bq * Hq);
    float* h0     = carve(sizeof(float) * (size_t)Bq * Hq);
    float* m_proj = carve(sizeof(float) * (size_t)Bq * Hq);
    float* m_cat  = carve(sizeof(float) * (size_t)Bq * 2 * Hq);
    float* bsum_p = carve(sizeof(float) * G4H);
    float* bsum_h = carve(sizeof(float) * G4H);
    float* bsum_m = carve(sizeof(float) * G4H);

    const int T = 256;

    // combined biases (bih + bhh) and zero initial match state
    vec_add2<<<(G4H + T - 1) / T, T, 0, stream>>>(bih_p, bhh_p, bsum_p, G4H);
    vec_add2<<<(G4H + T - 1) / T, T, 0, stream>>>(bih_h, bhh_h, bsum_h, G4H);
    vec_add2<<<(G4H + T - 1) / T, T, 0, stream>>>(bih_m, bhh_m, bsum_m, G4H);
    zero_f32<<<((Bq * Hq) + T - 1) / T, T, 0, stream>>>(h0, Bq * Hq);

    const int M = (int)tokrows;                        // 8192 tokens
    dim3 gGate((M + BMT - 1) / BMT, (G4H + BNT - 1) / BNT);   // 64 x 19
    dim3 gProj((M + BMT - 1) / BMT, (Hq + BNT - 1) / BNT);    // 64 x 5

    // premise token-LSTM (zero state): gates = X @ Wih_p^T + bsum ; h_s = act
    gather_embed<<<((M * Eq) + T - 1) / T, T, 0, stream>>>(premise, embed, xbuf, M);
    wmma_gemm_xwt<<<gGate, T, 0, stream>>>(xbuf, Wih_p, bsum_p, gates,
                                           M, G4H, Eq, Eq, Eq, G4H, 1);
    lstm_act<<<((M * Hq) + T - 1) / T, T, 0, stream>>>(gates, h_s, M);

    // hypothesis token-LSTM
    gather_embed<<<((M * Eq) + T - 1) / T, T, 0, stream>>>(hypothesis, embed, xbuf, M);
    wmma_gemm_xwt<<<gGate, T, 0, stream>>>(xbuf, Wih_h, bsum_h, gates,
                                           M, G4H, Eq, Eq, Eq, G4H, 1);
    lstm_act<<<((M * Hq) + T - 1) / T, T, 0, stream>>>(gates, h_t, M);

    // projections: s_proj = h_s @ W_s^T ; t_proj = h_t @ W_t^T
    wmma_gemm_xwt<<<gProj, T, 0, stream>>>(h_s, W_s, nullptr, s_proj,
                                           M, Hq, Hq, Hq, Hq, Hq, 0);
    wmma_gemm_xwt<<<gProj, T, 0, stream>>>(h_t, W_t, nullptr, t_proj,
                                           M, Hq, Hq, Hq, Hq, Hq, 0);

    // sequential match-LSTM scan over hypothesis steps
    dim3 gM((Bq + BMT - 1) / BMT, (Hq + BNT - 1) / BNT);      // 1 x 5
    dim3 gLM((Bq + BMT - 1) / BMT, (G4H + BNT - 1) / BNT);    // 1 x 19
    for (int k = 0; k < Kq; ++k) {
        const float* hprev = (k == 0) ? h0 : (hs_all + (size_t)(k - 1) * Bq * Hq);
        wmma_gemm_xwt<<<gM, T, 0, stream>>>(hprev, W_m, nullptr, m_proj,
                                            Bq, Hq, Hq, Hq, Hq, Hq, 0);
        attn_step<<<Bq, T, 0, stream>>>(s_proj, t_proj, m_proj, h_s, h_t,
                                        w_e, premise_len, m_cat, k);
        wmma_gemm_xwt<<<gLM, T, 0, stream>>>(m_cat, Wih_m, bsum_m, gates,
                                             Bq, G4H, 2 * Hq, 2 * Hq, 2 * Hq, G4H, 1);
        lstm_act<<<((Bq * Hq) + T - 1) / T, T, 0, stream>>>(
            gates, hs_all + (size_t)k * Bq * Hq, Bq);
    }

    // gather last valid h_m, FC + softmax
    final_softmax<<<Bq, 32, 0, stream>>>(hs_all, hypothesis_len, fc_w, fc_b,
                                         (float*)d_out);
}